// SimplePointPillars_81372450390075
// MI455X (gfx1250) — compile-verified
//
#include <hip/hip_runtime.h>

typedef __attribute__((ext_vector_type(2))) float v2f;
typedef __attribute__((ext_vector_type(4))) float v4f;
typedef __attribute__((ext_vector_type(8))) float v8f;

#define B_SZ      4
#define N_PTS     100000
#define TOT_PTS   (B_SZ * N_PTS)            // 400000
#define C_BEV     64
#define NXX       512
#define NYY       512
#define PLANE     (NYY * NXX)               // 262144
#define OUT_ELEMS (B_SZ * C_BEV * PLANE)    // 67108864
#define KILL_BIT  (1 << 30)
#define BASE_MASK ((1 << 27) - 1)           // base < 2^27

// ws layout (floats): [0..13] moment accumulators, [16..79] scale, [80..143] shift

__global__ __launch_bounds__(256) void k_zero(float* __restrict__ out, float* __restrict__ ws) {
    size_t i = (size_t)blockIdx.x * blockDim.x + threadIdx.x;
    size_t stride = (size_t)gridDim.x * blockDim.x;
    v4f z = {0.f, 0.f, 0.f, 0.f};
    v4f* o4 = (v4f*)out;
    for (size_t j = i; j < (size_t)(OUT_ELEMS / 4); j += stride) o4[j] = z;
    if (blockIdx.x == 0 && threadIdx.x < 16) ws[threadIdx.x] = 0.f;
}

// Reduce 4 first moments + 10 second moments of p over all points.
__global__ __launch_bounds__(256) void k_moments(const float* __restrict__ pts,
                                                 float* __restrict__ ws) {
    __shared__ float acc[14];
    if (threadIdx.x < 14) acc[threadIdx.x] = 0.f;
    __syncthreads();

    float s[4]   = {0.f, 0.f, 0.f, 0.f};
    float mm[10] = {0.f, 0.f, 0.f, 0.f, 0.f, 0.f, 0.f, 0.f, 0.f, 0.f};

    int i = blockIdx.x * blockDim.x + threadIdx.x;
    int stride = gridDim.x * blockDim.x;
    const v4f* p4 = (const v4f*)pts;
    for (; i < TOT_PTS; i += stride) {
        v4f p = p4[i];
        float v[4] = {p.x, p.y, p.z, p.w};
        int k = 0;
        #pragma unroll
        for (int a = 0; a < 4; ++a) {
            s[a] += v[a];
            #pragma unroll
            for (int c = a; c < 4; ++c) mm[k++] += v[a] * v[c];
        }
    }
    // wave32 tree reduce
    #pragma unroll
    for (int off = 16; off > 0; off >>= 1) {
        #pragma unroll
        for (int a = 0; a < 4; ++a)  s[a]  += __shfl_down(s[a],  off, 32);
        #pragma unroll
        for (int k = 0; k < 10; ++k) mm[k] += __shfl_down(mm[k], off, 32);
    }
    if ((threadIdx.x & 31) == 0) {
        #pragma unroll
        for (int a = 0; a < 4; ++a)  atomicAdd(&acc[a], s[a]);
        #pragma unroll
        for (int k = 0; k < 10; ++k) atomicAdd(&acc[4 + k], mm[k]);
    }
    __syncthreads();
    if (threadIdx.x < 14) atomicAdd(&ws[threadIdx.x], acc[threadIdx.x]);
}

// Per-channel affine fold: scale = gamma*rsqrt(var+eps), shift = beta + scale*(b - mu)
__global__ void k_finalize(const float* __restrict__ W, const float* __restrict__ bias,
                           const float* __restrict__ gamma, const float* __restrict__ beta,
                           float* __restrict__ ws) {
    int c = threadIdx.x;
    if (c >= C_BEV) return;
    const float inv = 1.0f / (float)TOT_PTS;
    float pb[4];
    #pragma unroll
    for (int a = 0; a < 4; ++a) pb[a] = ws[a] * inv;
    float wv[4];
    #pragma unroll
    for (int a = 0; a < 4; ++a) wv[a] = W[a * C_BEV + c];
    float bc = bias[c];
    float wdotp = wv[0]*pb[0] + wv[1]*pb[1] + wv[2]*pb[2] + wv[3]*pb[3];
    float mu = wdotp + bc;
    float E2 = 0.f;
    int k = 0;
    #pragma unroll
    for (int a = 0; a < 4; ++a)
        #pragma unroll
        for (int c2 = a; c2 < 4; ++c2) {
            float M = ws[4 + k] * inv; ++k;
            E2 += (a == c2 ? 1.f : 2.f) * wv[a] * wv[c2] * M;
        }
    E2 += 2.f * bc * wdotp + bc * bc;
    float var = E2 - mu * mu;
    float scale = gamma[c] * rsqrtf(var + 1e-5f);
    float shift = beta[c] + scale * (bc - mu);
    ws[16 + c] = scale;
    ws[80 + c] = shift;
}

// One wave = 16 points x 64 channels via 4x V_WMMA_F32_16X16X4_F32, then
// fused BN+ReLU and a fully branchless atomic scatter into (B,C,ny,nx).
// Invalid points are clamped to a border cell and contribute +0.0f (harmless:
// all accumulated values are >= 0 and the grid is +0.0-initialized).
__global__ __launch_bounds__(128) void k_scatter(const float* __restrict__ pts,
                                                 const float* __restrict__ W,
                                                 const float* __restrict__ ws,
                                                 float* __restrict__ out) {
    const int lane = threadIdx.x & 31;
    const int wave = threadIdx.x >> 5;
    const int tile = blockIdx.x * 4 + wave;       // 25000 tiles exactly
    const int m    = lane & 15;                   // point row within tile
    const int hi   = lane >> 4;                   // 0: K=0,1  1: K=2,3

    const int p = tile * 16 + m;                  // global point id
    v4f pt = ((const v4f*)pts)[p];

    // A operand: 16x4 f32, lanes 0-15 hold K=0(V0),K=1(V1); lanes 16-31 K=2,K=3
    v2f a;
    a.x = hi ? pt.z : pt.x;
    a.y = hi ? pt.w : pt.y;

    // Cell index (truncation toward zero matches astype(int32))
    int xc = (int)((pt.x + 51.2f) / 0.2f);
    int yc = (int)((pt.y + 51.2f) / 0.2f);
    int zc = (int)((pt.z + 10.0f) / 20.0f);
    bool kept = (xc >= 0) & (xc < NXX) & (yc >= 0) & (yc < NYY) & (zc == 0);
    int xcc = min(max(xc, 0), NXX - 1);
    int ycc = min(max(yc, 0), NYY - 1);
    int batch = p / N_PTS;
    // Packed: bits[26:0] = always-valid cell base, bit30 = kill flag
    int basep = (batch * (C_BEV * PLANE) + ycc * NXX + xcc) | (kept ? 0 : KILL_BIT);

    // B operand per channel-group g: 4x16 slice of W (mirror of A layout).
    v8f d[4];
    v8f cz = {0.f, 0.f, 0.f, 0.f, 0.f, 0.f, 0.f, 0.f};
    #pragma unroll
    for (int g = 0; g < 4; ++g) {
        int cidx = g * 16 + m;
        v2f bg;
        bg.x = W[(2 * hi)     * C_BEV + cidx];
        bg.y = W[(2 * hi + 1) * C_BEV + cidx];
        d[g] = __builtin_amdgcn_wmma_f32_16x16x4_f32(
            false, a, false, bg, (short)0, cz, false, false);
    }

    // D layout: lane l, VGPR r -> M = r + 8*hi, N = l&15. Broadcast packed
    // base of point M from lane M (single bpermute each).
    int rowbase[8];
    #pragma unroll
    for (int r = 0; r < 8; ++r) rowbase[r] = __shfl(basep, r + 8 * hi, 32);

    const float* scale = ws + 16;
    const float* shift = ws + 80;
    #pragma unroll
    for (int g = 0; g < 4; ++g) {
        int cidx = g * 16 + m;
        float sc = scale[cidx];
        float sh = shift[cidx];
        size_t coff = (size_t)cidx * PLANE;
        #pragma unroll
        for (int r = 0; r < 8; ++r) {
            int bp = rowbase[r];
            float v = fmaxf(d[g][r] * sc + sh, 0.f);   // BN + ReLU (b folded)
            v = (bp & KILL_BIT) ? 0.f : v;             // one cndmask, no branch
            atomicAdd(out + (size_t)(bp & BASE_MASK) + coff, v);
        }
    }
}

extern "C" void kernel_launch(void* const* d_in, const int* in_sizes, int n_in,
                              void* d_out, int out_size, void* d_ws, size_t ws_size,
                              hipStream_t stream) {
    const float* pts   = (const float*)d_in[0];
    const float* W     = (const float*)d_in[1];
    const float* bias  = (const float*)d_in[2];
    const float* gamma = (const float*)d_in[3];
    const float* beta  = (const float*)d_in[4];
    float* out = (float*)d_out;
    float* ws  = (float*)d_ws;

    k_zero<<<8192, 256, 0, stream>>>(out, ws);
    k_moments<<<512, 256, 0, stream>>>(pts, ws);
    k_finalize<<<1, 64, 0, stream>>>(W, bias, gamma, beta, ws);
    k_scatter<<<TOT_PTS / 64, 128, 0, stream>>>(pts, W, ws, out);  // 6250 blocks x 4 waves
}